// MultiLayerAttnDecoderRNNDot_7206955122724
// MI455X (gfx1250) — compile-verified
//
#include <hip/hip_runtime.h>
#include <math.h>

#define H 1024
#define V 128000
#define L_ENC 2048
#define N_LAYERS 5

typedef __attribute__((ext_vector_type(2))) float v2f;
typedef __attribute__((ext_vector_type(8))) float v8f;

// ---------------------------------------------------------------------------
// GEMV tile via V_WMMA_F32_16X16X4_F32.
// Each wave computes y[m0..m0+15] = W[m0..m0+15, :] . x  (+bias, opt relu).
// A (16x4 f32): lanes 0-15 hold M=0..15 {K=k,k+1}; lanes 16-31 {K=k+2,k+3}.
// B (4x16 f32): row K striped the same way; we broadcast x[k] across all 16
// columns, so every column of D equals the GEMV result. Lanes 0 and 16 hold
// column N=0 for rows m0..m0+7 / m0+8..m0+15 in acc[0..7].
// ---------------------------------------------------------------------------
__device__ __forceinline__ void gemv_tile_wmma(const float* __restrict__ W,
                                               const float* __restrict__ sx,
                                               const float* __restrict__ bias,
                                               float* __restrict__ y,
                                               int K, int m0, int relu) {
  const int lane = threadIdx.x & 31;
  const int half = lane >> 4;
  const int r    = lane & 15;
  const float* wrow = W + (size_t)(m0 + r) * (size_t)K + (half << 1);
  const float* xb   = sx + (half << 1);
  v8f acc = {};
#pragma unroll 4
  for (int k = 0; k < K; k += 4) {
    v2f a = *(const v2f*)(wrow + k);
    v2f b;
    b.x = xb[k];
    b.y = xb[k + 1];
    acc = __builtin_amdgcn_wmma_f32_16x16x4_f32(false, a, false, b,
                                                (short)0, acc, false, false);
  }
  if ((lane & 15) == 0) {  // lanes 0 and 16 carry column N=0
    const int base = m0 + (half << 3);
#pragma unroll
    for (int i = 0; i < 8; ++i) {
      float v = acc[i];
      if (bias) v += bias[base + i];
      if (relu) v = fmaxf(v, 0.0f);
      y[base + i] = v;
    }
  }
}

// Generic GEMV: grid.x * 8 waves * 16 rows = M. K <= 2048.
__global__ void gemv16_kernel(const float* __restrict__ W, const float* __restrict__ x,
                              const float* __restrict__ bias, float* __restrict__ y,
                              int K, int relu) {
  __shared__ float sx[2048];
  for (int i = threadIdx.x; i < K; i += blockDim.x) sx[i] = x[i];
  __syncthreads();
  const int wave = threadIdx.x >> 5;
  gemv_tile_wmma(W, sx, bias, y, K, (blockIdx.x * 8 + wave) * 16, relu);
}

// GRU gate GEMVs: blocks [0,24) -> gi = Wih.x + bih ; [24,48) -> gh = Whh.h + bhh
__global__ void gru_gemv_kernel(const float* __restrict__ Wih, const float* __restrict__ x,
                                const float* __restrict__ bih,
                                const float* __restrict__ Whh, const float* __restrict__ h,
                                const float* __restrict__ bhh,
                                float* __restrict__ gi, float* __restrict__ gh) {
  __shared__ float sx[H];
  const bool second = blockIdx.x >= 24;
  const float* W  = second ? Whh : Wih;
  const float* xv = second ? h   : x;
  const float* b  = second ? bhh : bih;
  float*       y  = second ? gh  : gi;
  const int blk   = second ? (int)blockIdx.x - 24 : (int)blockIdx.x;
  for (int i = threadIdx.x; i < H; i += blockDim.x) sx[i] = xv[i];
  __syncthreads();
  const int wave = threadIdx.x >> 5;
  gemv_tile_wmma(W, sx, b, y, H, (blk * 8 + wave) * 16, 0);
}

__global__ void embed_kernel(const float* __restrict__ emb, const int* __restrict__ token,
                             float* __restrict__ combined) {
  const int i = blockIdx.x * blockDim.x + threadIdx.x;
  combined[i] = emb[(size_t)token[0] * H + i];
}

__global__ void softmax2048_kernel(const float* __restrict__ scores, float* __restrict__ w) {
  __shared__ float sv[L_ENC];
  __shared__ float red[256];
  const int t = threadIdx.x;
  float mx = -INFINITY;
  for (int i = t; i < L_ENC; i += 256) { float v = scores[i]; sv[i] = v; mx = fmaxf(mx, v); }
  red[t] = mx; __syncthreads();
  for (int s = 128; s > 0; s >>= 1) { if (t < s) red[t] = fmaxf(red[t], red[t + s]); __syncthreads(); }
  mx = red[0]; __syncthreads();
  float sum = 0.0f;
  for (int i = t; i < L_ENC; i += 256) { float e = expf(sv[i] - mx); sv[i] = e; sum += e; }
  red[t] = sum; __syncthreads();
  for (int s = 128; s > 0; s >>= 1) { if (t < s) red[t] += red[t + s]; __syncthreads(); }
  const float inv = 1.0f / red[0];
  for (int i = t; i < L_ENC; i += 256) w[i] = sv[i] * inv;
}

// ctx = attn_weights @ enc  (2048x1024). 16 blocks x 64 cols, 4 j-groups/col.
__global__ void attn_apply_kernel(const float* __restrict__ enc, const float* __restrict__ w,
                                  float* __restrict__ out) {
  __shared__ float sw[L_ENC];
  __shared__ float sred[256];
  for (int i = threadIdx.x; i < L_ENC; i += blockDim.x) sw[i] = w[i];
  __syncthreads();
  const int c   = threadIdx.x & 63;
  const int jg  = threadIdx.x >> 6;   // 0..3
  const int col = blockIdx.x * 64 + c;
  float acc = 0.0f;
  for (int j = jg; j < L_ENC; j += 4)
    acc = fmaf(sw[j], enc[(size_t)j * H + col], acc);
  sred[threadIdx.x] = acc;
  __syncthreads();
  if (jg == 0)
    out[col] = (sred[c] + sred[c + 64]) + (sred[c + 128] + sred[c + 192]);
}

__global__ void gru_update_kernel(const float* __restrict__ gi, const float* __restrict__ gh,
                                  const float* __restrict__ h,
                                  float* __restrict__ x, float* __restrict__ hout) {
  const int i = blockIdx.x * blockDim.x + threadIdx.x;
  const float ir = gi[i], iz = gi[i + H], in_ = gi[i + 2 * H];
  const float hr = gh[i], hz = gh[i + H], hn  = gh[i + 2 * H];
  const float rg = 1.0f / (1.0f + expf(-(ir + hr)));
  const float zg = 1.0f / (1.0f + expf(-(iz + hz)));
  const float ng = tanhf(in_ + rg * hn);
  const float hnew = (1.0f - zg) * ng + zg * h[i];
  x[i] = hnew;
  hout[i] = hnew;
}

// single-block log-sum-exp over the 128000 logits
__global__ void lse_kernel(const float* __restrict__ logits, float* __restrict__ red) {
  __shared__ float s[1024];
  const int t = threadIdx.x;
  float mx = -INFINITY;
  for (int i = t; i < V; i += 1024) mx = fmaxf(mx, logits[i]);
  s[t] = mx; __syncthreads();
  for (int st = 512; st > 0; st >>= 1) { if (t < st) s[t] = fmaxf(s[t], s[t + st]); __syncthreads(); }
  mx = s[0]; __syncthreads();
  float sum = 0.0f;
  for (int i = t; i < V; i += 1024) sum += expf(logits[i] - mx);
  s[t] = sum; __syncthreads();
  for (int st = 512; st > 0; st >>= 1) { if (t < st) s[t] += s[t + st]; __syncthreads(); }
  if (t == 0) red[0] = mx + logf(s[0]);
}

__global__ void finalize_kernel(float* __restrict__ logits, const float* __restrict__ red) {
  const int i = blockIdx.x * blockDim.x + threadIdx.x;
  logits[i] -= red[0];
}

extern "C" void kernel_launch(void* const* d_in, const int* in_sizes, int n_in,
                              void* d_out, int out_size, void* d_ws, size_t ws_size,
                              hipStream_t stream) {
  const int*   token  = (const int*)  d_in[0];
  const float* hidden = (const float*)d_in[1];   // (5,1,1024)
  const float* enc    = (const float*)d_in[2];   // (2048,1024)
  const float* emb    = (const float*)d_in[3];   // (128000,1024)
  const float* Wc     = (const float*)d_in[4];   // (1024,2048)
  const float* bc     = (const float*)d_in[5];
  const float* Wih    = (const float*)d_in[6];   // (5,3072,1024)
  const float* Whh    = (const float*)d_in[7];
  const float* bih    = (const float*)d_in[8];   // (5,3072)
  const float* bhh    = (const float*)d_in[9];
  const float* Wout   = (const float*)d_in[10];  // (128000,1024)
  const float* bout   = (const float*)d_in[11];

  float* out      = (float*)d_out;
  float* logp     = out;                    // [128000] (logits in-place, then corrected)
  float* hid_out  = out + V;                // [5*1024]
  float* attn_out = out + V + N_LAYERS * H; // [2048]

  float* ws       = (float*)d_ws;
  float* scores   = ws;            // 2048
  float* combined = ws + 2048;     // 2048
  float* xbuf     = ws + 4096;     // 1024
  float* gi       = ws + 5120;     // 3072
  float* gh       = ws + 8192;     // 3072
  float* red      = ws + 11264;    // 1

  // embedding row + attention scores (independent)
  embed_kernel<<<4, 256, 0, stream>>>(emb, token, combined);
  gemv16_kernel<<<L_ENC / 128, 256, 0, stream>>>(enc, hidden, nullptr, scores, H, 0);
  // softmax -> attn_weights (final output, also consumed below)
  softmax2048_kernel<<<1, 256, 0, stream>>>(scores, attn_out);
  // context vector into combined[1024..2047]
  attn_apply_kernel<<<16, 256, 0, stream>>>(enc, attn_out, combined + H);
  // x = relu(W_combine . combined + b)
  gemv16_kernel<<<H / 128, 256, 0, stream>>>(Wc, combined, bc, xbuf, 2 * H, 1);
  // 5 GRU layers
  for (int l = 0; l < N_LAYERS; ++l) {
    const float* hl = hidden + (size_t)l * H;
    gru_gemv_kernel<<<48, 256, 0, stream>>>(Wih + (size_t)l * 3 * H * H, xbuf,
                                            bih + (size_t)l * 3 * H,
                                            Whh + (size_t)l * 3 * H * H, hl,
                                            bhh + (size_t)l * 3 * H, gi, gh);
    gru_update_kernel<<<4, 256, 0, stream>>>(gi, gh, hl, xbuf, hid_out + (size_t)l * H);
  }
  // logits = W_out . x + b_out, then log-softmax in place
  gemv16_kernel<<<V / 128, 256, 0, stream>>>(Wout, xbuf, bout, logp, H, 0);
  lse_kernel<<<1, 1024, 0, stream>>>(logp, red);
  finalize_kernel<<<V / 256, 256, 0, stream>>>(logp, red);
}